// EnrichedGeometricEmbedding_21397527069083
// MI455X (gfx1250) — compile-verified
//
#include <hip/hip_runtime.h>

// ---------------------------------------------------------------------------
// EnrichedGeometricEmbedding on MI455X (gfx1250, wave32)
//   pass 0: pre-transpose W into Wt[384][140] (zero K-padding baked in)
//   pass 1: per-group 3x3 PCA stats (mean + curvature) via wave shuffles
//   pass 2: fused feature build + fp32 WMMA GEMM, weight strips streamed
//           into double-buffered LDS by the Tensor Data Mover (TENSORcnt),
//           bias preloaded per strip, non-temporal output stores
// ---------------------------------------------------------------------------

typedef __attribute__((ext_vector_type(2))) float v2f;
typedef __attribute__((ext_vector_type(8))) float v8f;
typedef __attribute__((ext_vector_type(4))) unsigned int u32x4;
typedef __attribute__((ext_vector_type(8))) int i32x8;
typedef __attribute__((ext_vector_type(4))) int i32x4;

#define IN_DIM     3
#define FEAT_DIM   43
#define RBF_DIM    (IN_DIM * FEAT_DIM)      // 129
#define TOTAL_DIM  133
#define NCHUNK     34                        // K padded to 136 = 34 * 4
#define OUT_DIM    384
#define KPTS       32
#define NSTRIP     96                        // N columns staged in LDS per strip
#define CSTRIDE    140                       // padded c-stride (>= 136, even)
#define NTILES     (NSTRIP / 16)             // 6 output tiles per strip
#define NSTRIPS    (OUT_DIM / NSTRIP)        // 4 strips

// ------------------------------ pass 0 -------------------------------------
// Wt[n * CSTRIDE + c] = W[c][n]  (c >= TOTAL_DIM zero-padded)
__global__ __launch_bounds__(256)
void transpose_w_kernel(const float* __restrict__ W, float* __restrict__ Wt) {
  const int e = blockIdx.x * 256 + threadIdx.x;
  if (e >= OUT_DIM * CSTRIDE) return;
  const int n = e / CSTRIDE;
  const int c = e - n * CSTRIDE;
  Wt[e] = (c < TOTAL_DIM) ? W[(size_t)c * OUT_DIM + n] : 0.0f;
}

// ------------------------------ pass 1 -------------------------------------
__global__ __launch_bounds__(256)
void pca_stats_kernel(const float* __restrict__ nxyz,
                      float* __restrict__ stats /* [G]{mx,my,mz,curv} */) {
  const int wave = (blockIdx.x * blockDim.x + threadIdx.x) >> 5;  // group id
  const int lane = threadIdx.x & 31;

  const float* p = nxyz + (size_t)wave * (KPTS * 3) + lane * 3;
  float x = __builtin_nontemporal_load(p + 0);   // stream-once input
  float y = __builtin_nontemporal_load(p + 1);
  float z = __builtin_nontemporal_load(p + 2);

  float sx = x, sy = y, sz = z;
  #pragma unroll
  for (int m = 16; m > 0; m >>= 1) {
    sx += __shfl_xor(sx, m, 32);
    sy += __shfl_xor(sy, m, 32);
    sz += __shfl_xor(sz, m, 32);
  }
  const float mx = sx * (1.0f / 32.0f);
  const float my = sy * (1.0f / 32.0f);
  const float mz = sz * (1.0f / 32.0f);

  const float cx = x - mx, cy = y - my, cz = z - mz;
  float xx = cx * cx, xy = cx * cy, xz = cx * cz;
  float yy = cy * cy, yz = cy * cz, zz = cz * cz;
  #pragma unroll
  for (int m = 16; m > 0; m >>= 1) {
    xx += __shfl_xor(xx, m, 32);
    xy += __shfl_xor(xy, m, 32);
    xz += __shfl_xor(xz, m, 32);
    yy += __shfl_xor(yy, m, 32);
    yz += __shfl_xor(yz, m, 32);
    zz += __shfl_xor(zz, m, 32);
  }
  const float s = 1.0f / (KPTS - 1);
  xx *= s; xy *= s; xz *= s; yy *= s; yz *= s; zz *= s;

  const float tr = xx + yy + zz;
  const float q  = tr * (1.0f / 3.0f);
  const float p1 = xy * xy + xz * xz + yz * yz;
  const float p2 = (xx - q) * (xx - q) + (yy - q) * (yy - q) +
                   (zz - q) * (zz - q) + 2.0f * p1;
  float curv;
  if (p2 < 1e-22f) {
    curv = q / (tr + 1e-6f);
  } else {
    const float pp  = sqrtf(p2 * (1.0f / 6.0f));
    const float inv = 1.0f / pp;
    const float bxx = (xx - q) * inv, byy = (yy - q) * inv, bzz = (zz - q) * inv;
    const float bxy = xy * inv, bxz = xz * inv, byz = yz * inv;
    float detB = bxx * (byy * bzz - byz * byz)
               - bxy * (bxy * bzz - byz * bxz)
               + bxz * (bxy * byz - byy * bxz);
    float r = fminf(1.0f, fmaxf(-1.0f, 0.5f * detB));
    const float phi = acosf(r) * (1.0f / 3.0f);
    const float emin = q + 2.0f * pp * cosf(phi + 2.0943951023931953f);
    curv = emin / (tr + 1e-6f);
  }

  if (lane == 0) {
    float4 st;
    st.x = mx; st.y = my; st.z = mz; st.w = curv;
    *(float4*)(stats + (size_t)wave * 4) = st;
  }
}

// ---------------------- TDM strip DMA (Wt tile -> LDS) ----------------------
// 2-D tile: tile_dim0 = CSTRIDE (contiguous floats), tile_dim1 = NSTRIP rows,
// row stride = CSTRIDE. Lands in LDS exactly in the compute layout.
__device__ __forceinline__ void tdm_load_strip(const float* gsrc,
                                               unsigned lds_byte_off) {
  const unsigned long long ga = (unsigned long long)(size_t)gsrc;
  u32x4 g0;
  g0[0] = 1u;                                            // count=1, user mode
  g0[1] = lds_byte_off;                                  // lds_addr (bytes)
  g0[2] = (unsigned)ga;                                  // global_addr[31:0]
  g0[3] = (unsigned)((ga >> 32) & 0x01FFFFFFull) | 0x80000000u; // [56:32]|type=2

  i32x8 g1;
  g1[0] = (int)(2u << 16);                               // data_size = 4 bytes
  g1[1] = (int)((unsigned)CSTRIDE << 16);                // tensor_dim0[15:0]
  g1[2] = (int)((unsigned)NSTRIP << 16);                 // dim0 hi=0 | dim1 lo
  g1[3] = (int)((unsigned)CSTRIDE << 16);                // dim1 hi=0 | tile_dim0
  g1[4] = (int)(unsigned)NSTRIP;                         // tile_dim1 | tile_dim2=0
  g1[5] = (int)(unsigned)CSTRIDE;                        // tensor_dim0_stride lo
  g1[6] = 0;                                             // stride0 hi | stride1 lo
  g1[7] = 0;                                             // stride1 hi

  i32x4 z4 = {0, 0, 0, 0};
#if defined(__clang_major__) && (__clang_major__ >= 23)
  i32x8 z8 = {0, 0, 0, 0, 0, 0, 0, 0};
  __builtin_amdgcn_tensor_load_to_lds(g0, g1, z4, z4, z8, 0);
#else
  __builtin_amdgcn_tensor_load_to_lds(g0, g1, z4, z4, 0);
#endif
}

// ------------------------------ pass 2 -------------------------------------
__global__ __launch_bounds__(256)
void embed_gemm_kernel(const float* __restrict__ xyz,
                       const float* __restrict__ Wt,     // [384, CSTRIDE]
                       const float* __restrict__ bias,   // [384]
                       const float* __restrict__ stats,  // [G]{mx,my,mz,curv}
                       float* __restrict__ out) {        // [M, 384]
  __shared__ float sW[2][NSTRIP * CSTRIDE];              // 2 x 53.76 KB

  const int t    = threadIdx.x;
  const int lane = t & 31;
  const int wave = t >> 5;
  const int m    = lane & 15;    // matrix row within the 16-row tile
  const int half = lane >> 4;    // which K-pair this lane feeds (WMMA layout)

  const int rowBase = blockIdx.x * 128 + wave * 16;      // aligned: one group
  const int r = rowBase + m;
  const int g = r >> 5;

  const float x0 = __builtin_nontemporal_load(xyz + 3 * (size_t)r + 0);
  const float x1 = __builtin_nontemporal_load(xyz + 3 * (size_t)r + 1);
  const float x2 = __builtin_nontemporal_load(xyz + 3 * (size_t)r + 2);
  const float4 st = *(const float4*)(stats + 4 * (size_t)g);

  // Build all 34 A-fragments (16x4 f32) in registers, fused features:
  // lane layout per ISA: v0 = A[m][4kc + 2*half], v1 = A[m][4kc + 2*half + 1]
  v2f afrag[NCHUNK];
  const float step = 2.0f / (FEAT_DIM + 1);
  #pragma unroll
  for (int kc = 0; kc < NCHUNK; ++kc) {
    float vv[2];
    #pragma unroll
    for (int j = 0; j < 2; ++j) {
      const int c = 4 * kc + 2 * half + j;
      float val;
      if (c < RBF_DIM) {
        const int d = c / FEAT_DIM;
        const int f = c - d * FEAT_DIM;
        const float xd = (d == 0) ? x0 : ((d == 1) ? x1 : x2);
        const float fv = -1.0f + (float)(f + 1) * step;
        const float tt = xd - fv;
        val = __expf(-2.0f * tt * tt);                   // -0.5/sigma^2, sigma=0.5
      } else if (c == RBF_DIM) {
        val = st.w;                                      // curvature cue
      } else if (c < TOTAL_DIM) {
        const int d = c - RBF_DIM - 1;
        const float xd = (d == 0) ? x0 : ((d == 1) ? x1 : x2);
        const float md = (d == 0) ? st.x : ((d == 1) ? st.y : st.z);
        val = fabsf(xd - md);                            // laplacian cue
      } else {
        val = 0.0f;                                      // K padding 133..135
      }
      vv[j] = val;
    }
    v2f a; a.x = vv[0]; a.y = vv[1];
    afrag[kc] = a;
  }

  // LDS byte offsets of the two strip buffers (addrspace(3) -> raw offset)
  typedef __attribute__((address_space(3))) float lds_f;
  const unsigned ldsOff0 = (unsigned)(unsigned long long)(lds_f*)&sW[0][0];
  const unsigned ldsOff1 = (unsigned)(unsigned long long)(lds_f*)&sW[1][0];

  // Prologue: wave 0 kicks off the TDM DMA for strip 0
  if (wave == 0) {
    tdm_load_strip(Wt, ldsOff0);
  }

  for (int strip = 0; strip < NSTRIPS; ++strip) {
    const int nbase = strip * NSTRIP;
    const int buf = strip & 1;

    // Preload this strip's bias values; latency hides behind the barrier.
    float bv[NTILES];
    #pragma unroll
    for (int i = 0; i < NTILES; ++i) {
      bv[i] = bias[nbase + i * 16 + m];
    }

    if (wave == 0) {
      __builtin_amdgcn_s_wait_tensorcnt(0);   // current strip DMA complete
    }
    __syncthreads();                          // publish LDS to all waves;
                                              // also: all reads of buf^1 done
    if (wave == 0 && strip + 1 < NSTRIPS) {
      tdm_load_strip(Wt + (size_t)(strip + 1) * NSTRIP * CSTRIDE,
                     buf ? ldsOff0 : ldsOff1);  // prefetch next strip
    }

    #pragma unroll
    for (int nt = 0; nt < NTILES; ++nt) {
      const int n = nbase + nt * 16 + m;
      const float b0 = bv[nt];
      v8f acc = {b0, b0, b0, b0, b0, b0, b0, b0};        // bias folded into C

      const float* bbase = &sW[buf][(nt * 16 + m) * CSTRIDE + 2 * half];
      #pragma unroll
      for (int kc = 0; kc < NCHUNK; ++kc) {
        const v2f b = *(const v2f*)(bbase + 4 * kc);     // ds_load_2addr_b64
        acc = __builtin_amdgcn_wmma_f32_16x16x4_f32(
            /*neg_a=*/false, afrag[kc], /*neg_b=*/false, b,
            /*c_mod=*/(short)0, acc, /*reuse_a=*/false, /*reuse_b=*/false);
      }

      // D layout: VGPR vr -> row (vr + 8*half), col n
      // Output is write-once and 2x the L2 -> stream with non-temporal stores.
      float* op = out + (size_t)(rowBase + 8 * half) * OUT_DIM + n;
      #pragma unroll
      for (int vr = 0; vr < 8; ++vr) {
        __builtin_nontemporal_store(acc[vr], op + (size_t)vr * OUT_DIM);
      }
    }
  }
}

// ------------------------------ launch -------------------------------------
extern "C" void kernel_launch(void* const* d_in, const int* in_sizes, int n_in,
                              void* d_out, int out_size, void* d_ws, size_t ws_size,
                              hipStream_t stream) {
  const float* xyz  = (const float*)d_in[0];
  const float* nxyz = (const float*)d_in[1];
  const float* W    = (const float*)d_in[2];
  const float* bias = (const float*)d_in[3];
  float* out = (float*)d_out;

  const int rows    = in_sizes[0] / IN_DIM;               // B*S*K = 262144
  const int ngroups = rows / KPTS;                        // 8192

  float* stats = (float*)d_ws;                            // ngroups * 4 floats
  float* Wt    = stats + (size_t)ngroups * 4;             // 384 * 140 floats

  // pass 0: transpose+pad W once (tiny; L2 resident afterwards)
  transpose_w_kernel<<<(OUT_DIM * CSTRIDE + 255) / 256, 256, 0, stream>>>(W, Wt);

  // pass 1: one wave32 per group
  pca_stats_kernel<<<(ngroups * 32) / 256, 256, 0, stream>>>(nxyz, stats);

  // pass 2: 128 rows per 256-thread block (8 waves x 16-row M tiles)
  embed_gemm_kernel<<<rows / 128, 256, 0, stream>>>(xyz, Wt, bias, stats, out);
}